// InteractionLayerV0_65730179498523
// MI455X (gfx1250) — compile-verified
//
#include <hip/hip_runtime.h>
#include <math.h>

#define E_EDGES 65536
#define N_NODES 8192
#define MUL 32
#define LAT 128
#define CONDN 32

typedef float v2f __attribute__((ext_vector_type(2)));
typedef float v8f __attribute__((ext_vector_type(8)));

__device__ __forceinline__ float wred_sum(float v) {
#pragma unroll
  for (int o = 16; o > 0; o >>= 1) v += __shfl_xor(v, o, 32);
  return v;
}

// ---- kernel 0a: zero node accumulator ----
__global__ __launch_bounds__(256) void k_zero(float* p, int n) {
  int i = blockIdx.x * 256 + threadIdx.x;
  if (i < n) p[i] = 0.f;
}

// ---- kernel 0b: residual base: out = co * [scalar | equiv] ----
__global__ __launch_bounds__(256) void k_resbase(const float* __restrict__ s,
                                                 const float* __restrict__ eq,
                                                 const float* __restrict__ uc,
                                                 float* __restrict__ out) {
  int i = blockIdx.x * 256 + threadIdx.x;
  float c = uc[0];
  float co = rsqrtf(c * c + 1.f);
  const int ns = E_EDGES * LAT;
  const int total = ns + E_EDGES * MUL * 9;
  if (i < ns) out[i] = co * s[i];
  else if (i < total) out[i] = co * eq[i - ns];
}

// ---- kernel 1: env channel-mix + gates + atomic scatter to nodes ----
__global__ __launch_bounds__(256) void k_env_scatter(
    const float* __restrict__ scal, const float* __restrict__ eq,
    const float* __restrict__ cond, const float* __restrict__ W_env,
    const float* __restrict__ b_env0, const float* __restrict__ Wg_env,
    const float* __restrict__ bg_env, const float* __restrict__ Wg_eq,
    const float* __restrict__ bg_eq, const int* __restrict__ edge_index,
    float* __restrict__ env_nodes, float* __restrict__ g_eq_out) {
  __shared__ float Ws[3 * MUL * MUL];
  __shared__ float xs[8][MUL][9];
  const int tid = threadIdx.x, wave = tid >> 5, lane = tid & 31;
  for (int i = tid; i < 3 * MUL * MUL; i += 256) Ws[i] = W_env[i];
  const int e = blockIdx.x * 8 + wave;
  const float* xp = eq + (size_t)e * (MUL * 9) + lane * 9;
#pragma unroll
  for (int m = 0; m < 9; m++) xs[wave][lane][m] = xp[m];
  float a_env = 0.f, a_eq = 0.f;
#pragma unroll
  for (int j = 0; j < 4; j++) {
    float v = scal[(size_t)e * LAT + lane + 32 * j];
    a_env += v * Wg_env[lane + 32 * j];
    a_eq  += v * Wg_eq [lane + 32 * j];
  }
  {
    float v = cond[(size_t)e * CONDN + lane];
    a_env += v * Wg_env[LAT + lane];
    a_eq  += v * Wg_eq [LAT + lane];
  }
  a_env = wred_sum(a_env);
  a_eq  = wred_sum(a_eq);
  float g_env = 1.f + 0.1f * tanhf(a_env + bg_env[0]);
  float g_eq  = 1.f + 0.1f * tanhf(a_eq  + bg_eq[0]);
  if (lane == 0) g_eq_out[e] = g_eq;
  __syncthreads();
  float acc[9];
#pragma unroll
  for (int m = 0; m < 9; m++) acc[m] = 0.f;
  const float* W0 = Ws;
  const float* W1 = Ws + MUL * MUL;
  const float* W2 = Ws + 2 * MUL * MUL;
  for (int u = 0; u < MUL; u++) {
    float w0 = W0[lane * MUL + u], w1 = W1[lane * MUL + u], w2 = W2[lane * MUL + u];
    acc[0] += w0 * xs[wave][u][0];
#pragma unroll
    for (int m = 1; m < 4; m++) acc[m] += w1 * xs[wave][u][m];
#pragma unroll
    for (int m = 4; m < 9; m++) acc[m] += w2 * xs[wave][u][m];
  }
  acc[0] += b_env0[lane];
  const int ctr = edge_index[e];  // row 0 of (2,E)
  float* dst = env_nodes + (size_t)ctr * (MUL * 9) + lane * 9;
#pragma unroll
  for (int m = 0; m < 9; m++) atomicAdd(&dst[m], acc[m] * g_env);
}

// ---- kernel 2: per-node SO(3) layernorm ----
__global__ __launch_bounds__(256) void k_node_ln(float* __restrict__ env_nodes,
                                                 const float* __restrict__ gamma_env) {
  const int tid = threadIdx.x, wave = tid >> 5, lane = tid & 31;
  const int n = blockIdx.x * 8 + wave;
  float* p = env_nodes + (size_t)n * (MUL * 9) + lane * 9;
  float x[9];
#pragma unroll
  for (int m = 0; m < 9; m++) x[m] = p[m];
  const float eps = 1e-6f;
  float mu = wred_sum(x[0]) * (1.f / 32.f);
  float d = x[0] - mu;
  float var = wred_sum(d * d) * (1.f / 32.f);
  x[0] = d * rsqrtf(var + eps) * gamma_env[lane];
  float ss = x[1] * x[1] + x[2] * x[2] + x[3] * x[3];
  float ms = wred_sum(ss) * (1.f / 96.f);
  float sc = rsqrtf(ms + eps) * gamma_env[32 + lane];
#pragma unroll
  for (int m = 1; m < 4; m++) x[m] *= sc;
  ss = 0.f;
#pragma unroll
  for (int m = 4; m < 9; m++) ss += x[m] * x[m];
  ms = wred_sum(ss) * (1.f / 160.f);
  sc = rsqrtf(ms + eps) * gamma_env[64 + lane];
#pragma unroll
  for (int m = 4; m < 9; m++) x[m] *= sc;
#pragma unroll
  for (int m = 0; m < 9; m++) p[m] = x[m];
}

// l=2 components <-> symmetric traceless 3x3 (basis: xy, yz, z2, xz, x2-y2).
// Per-path global scale is irrelevant: every TP path feeds a scale-invariant LN.
__device__ __forceinline__ void q_to_mat(const float* q, float M[3][3]) {
  const float is3 = 0.57735027f;
  M[0][0] = q[4] - q[2] * is3; M[0][1] = q[0];               M[0][2] = q[3];
  M[1][0] = q[0];              M[1][1] = -q[4] - q[2] * is3; M[1][2] = q[1];
  M[2][0] = q[3];              M[2][1] = q[1];               M[2][2] = 2.f * q[2] * is3;
}
__device__ __forceinline__ void sym_to_q(float Sxx, float Syy, float Szz,
                                         float Sxy, float Syz, float Sxz,
                                         float tr, float* q) {
  q[0] = Sxy;
  q[1] = Syz;
  q[3] = Sxz;
  q[4] = 0.5f * (Sxx - Syy);
  q[2] = 0.8660254f * (Szz - tr * (1.f / 3.f));
}

// WMMA epilogue for equivariant mixing: D rows = 16 edges, cols = 16 channels.
__device__ __forceinline__ void eq_epilogue(v8f acc, int e0, int nt, int comp,
                                            const float* __restrict__ g_eq,
                                            const int* __restrict__ active,
                                            float cn, float* __restrict__ out_e,
                                            int hi, int ln) {
  const int ch = nt * 16 + ln;
#pragma unroll
  for (int i = 0; i < 8; i++) {
    const int er = e0 + i + 8 * hi;
    const float g = g_eq[er];
    const int a = active[er];
    out_e[(size_t)a * (MUL * 9) + ch * 9 + comp] += cn * g * acc[i];
  }
}

// LDS strides padded for bank-conflict-free 16-lane A-fragment reads.
#define TPS_STRIDE 100   // 96 + 4
#define TL_STRIDE  196   // 192 + 4
#define TL_PLANE   (16 * TL_STRIDE)  // 3136 floats per m/k component plane

// ---- kernel 3a: tensor product + per-path LN + WMMA equivariant mixing ----
__global__ __launch_bounds__(512) void k_tp_mix(
    const float* __restrict__ eq, const float* __restrict__ env_nodes,
    const float* __restrict__ cond, const int* __restrict__ edge_index,
    const float* __restrict__ gamma_tp, const float* __restrict__ Weq0,
    const float* __restrict__ beq0, const float* __restrict__ Weq1,
    const float* __restrict__ Weq2, const float* __restrict__ g_eq_in,
    const float* __restrict__ uc, const int* __restrict__ active,
    float* __restrict__ sin_ws, float* __restrict__ out_e) {
  // phase A: [0,1600) tps[e][96+pad], [1600, 1600+3*3136) tl1[m][e][192+pad]
  // phase B: [0, 5*3136) tl2[k][e][192+pad]   (reused)
  __shared__ float buf[5 * TL_PLANE];  // 15680 floats = 62720 B
  float* tps = buf;
  float* tl1 = buf + 16 * TPS_STRIDE;

  const int tid = threadIdx.x, wave = tid >> 5, lane = tid & 31;
  const int hi = lane >> 4, ln = lane & 15;
  const int e0 = blockIdx.x * 16;
  const int e = e0 + wave;
  const float eps = 1e-6f;

  // ================= phase 1: per-edge TP + LN (one wave per edge) ==========
  float x1[9], x2[9];
  const float* p1 = eq + (size_t)e * (MUL * 9) + lane * 9;
  const int ctr = edge_index[e];
  const float* p2 = env_nodes + (size_t)ctr * (MUL * 9) + lane * 9;
#pragma unroll
  for (int m = 0; m < 9; m++) { x1[m] = p1[m]; x2[m] = p2[m]; }

  const float s1 = x1[0], s2 = x2[0];
  const float v1[3] = {x1[1], x1[2], x1[3]};
  const float v2[3] = {x2[1], x2[2], x2[3]};
  float M1[3][3], M2[3][3];
  q_to_mat(x1 + 4, M1);
  q_to_mat(x2 + 4, M2);
  float A[3][3];
#pragma unroll
  for (int i = 0; i < 3; i++)
#pragma unroll
    for (int j = 0; j < 3; j++)
      A[i][j] = M1[i][0] * M2[0][j] + M1[i][1] * M2[1][j] + M1[i][2] * M2[2][j];
  const float trA = A[0][0] + A[1][1] + A[2][2];

  float pl0[3];
  pl0[0] = s1 * s2;
  pl0[1] = v1[0] * v2[0] + v1[1] * v2[1] + v1[2] * v2[2];
  pl0[2] = trA;

  float pl1[6][3];
#pragma unroll
  for (int m = 0; m < 3; m++) { pl1[0][m] = s1 * v2[m]; pl1[1][m] = v1[m] * s2; }
  pl1[2][0] = v1[1] * v2[2] - v1[2] * v2[1];
  pl1[2][1] = v1[2] * v2[0] - v1[0] * v2[2];
  pl1[2][2] = v1[0] * v2[1] - v1[1] * v2[0];
#pragma unroll
  for (int i = 0; i < 3; i++) {
    pl1[3][i] = M2[i][0] * v1[0] + M2[i][1] * v1[1] + M2[i][2] * v1[2];
    pl1[4][i] = M1[i][0] * v2[0] + M1[i][1] * v2[1] + M1[i][2] * v2[2];
  }
  pl1[5][0] = A[1][2] - A[2][1];
  pl1[5][1] = A[2][0] - A[0][2];
  pl1[5][2] = A[0][1] - A[1][0];

  float pl2[6][5];
#pragma unroll
  for (int k = 0; k < 5; k++) { pl2[0][k] = s1 * x2[4 + k]; pl2[3][k] = x1[4 + k] * s2; }
  {
    float tr = 2.f * pl0[1];
    sym_to_q(2.f * v1[0] * v2[0], 2.f * v1[1] * v2[1], 2.f * v1[2] * v2[2],
             v1[0] * v2[1] + v1[1] * v2[0], v1[1] * v2[2] + v1[2] * v2[1],
             v1[0] * v2[2] + v1[2] * v2[0], tr, pl2[1]);
  }
  {
    float P[3][3];
#pragma unroll
    for (int j = 0; j < 3; j++) {
      P[0][j] = -v1[2] * M2[1][j] + v1[1] * M2[2][j];
      P[1][j] =  v1[2] * M2[0][j] - v1[0] * M2[2][j];
      P[2][j] = -v1[1] * M2[0][j] + v1[0] * M2[1][j];
    }
    sym_to_q(P[0][0], P[1][1], P[2][2], 0.5f * (P[0][1] + P[1][0]),
             0.5f * (P[1][2] + P[2][1]), 0.5f * (P[0][2] + P[2][0]), 0.f, pl2[2]);
  }
  {
    float P[3][3];
#pragma unroll
    for (int j = 0; j < 3; j++) {
      P[0][j] = -v2[2] * M1[1][j] + v2[1] * M1[2][j];
      P[1][j] =  v2[2] * M1[0][j] - v2[0] * M1[2][j];
      P[2][j] = -v2[1] * M1[0][j] + v2[0] * M1[1][j];
    }
    sym_to_q(P[0][0], P[1][1], P[2][2], 0.5f * (P[0][1] + P[1][0]),
             0.5f * (P[1][2] + P[2][1]), 0.5f * (P[0][2] + P[2][0]), 0.f, pl2[4]);
  }
  sym_to_q(A[0][0], A[1][1], A[2][2], 0.5f * (A[0][1] + A[1][0]),
           0.5f * (A[1][2] + A[2][1]), 0.5f * (A[0][2] + A[2][0]), trA, pl2[5]);

  // per-path LN; l=0 scalars in reference order u*3+p (concat axis=-1 + reshape)
#pragma unroll
  for (int p = 0; p < 3; p++) {
    float s = wred_sum(pl0[p]);
    float sq = wred_sum(pl0[p] * pl0[p]);
    float mu = s * (1.f / 32.f);
    float var = sq * (1.f / 32.f) - mu * mu;
    float v = (pl0[p] - mu) * rsqrtf(var + eps) * gamma_tp[p * 32 + lane];
    tps[wave * TPS_STRIDE + lane * 3 + p] = v;
    sin_ws[(size_t)e * LAT + lane * 3 + p] = v;
  }
  sin_ws[(size_t)e * LAT + 96 + lane] = cond[(size_t)e * CONDN + lane];
  float l2v[6][5];
#pragma unroll
  for (int p = 0; p < 6; p++) {
    float ss = pl1[p][0] * pl1[p][0] + pl1[p][1] * pl1[p][1] + pl1[p][2] * pl1[p][2];
    float ms = wred_sum(ss) * (1.f / 96.f);
    float sc = rsqrtf(ms + eps) * gamma_tp[(3 + p) * 32 + lane];
#pragma unroll
    for (int m = 0; m < 3; m++)
      tl1[m * TL_PLANE + wave * TL_STRIDE + p * 32 + lane] = pl1[p][m] * sc;
  }
#pragma unroll
  for (int p = 0; p < 6; p++) {
    float ss = 0.f;
#pragma unroll
    for (int k = 0; k < 5; k++) ss += pl2[p][k] * pl2[p][k];
    float ms = wred_sum(ss) * (1.f / 160.f);
    float sc = rsqrtf(ms + eps) * gamma_tp[(9 + p) * 32 + lane];
#pragma unroll
    for (int k = 0; k < 5; k++) l2v[p][k] = pl2[p][k] * sc;  // keep in regs
  }
  __syncthreads();

  const float c = uc[0];
  const float cn = c * rsqrtf(c * c + 1.f);

  // ============ phase A mixing: l=1 (6 jobs) + l=0 (2 jobs) via WMMA =========
  if (wave < 8) {
    if (wave < 6) {
      const int m = wave >> 1, nt = wave & 1;
      const float* Ab = tl1 + m * TL_PLANE;
      v8f acc;
#pragma unroll
      for (int i = 0; i < 8; i++) acc[i] = 0.f;
      for (int kk = 0; kk < 48; kk++) {
        const int k0 = kk * 4 + 2 * hi;
        float2 af = *(const float2*)&Ab[ln * TL_STRIDE + k0];
        float2 bf = *(const float2*)&Weq1[(size_t)(nt * 16 + ln) * 192 + k0];
        v2f a; a.x = af.x; a.y = af.y;
        v2f b; b.x = bf.x; b.y = bf.y;
        acc = __builtin_amdgcn_wmma_f32_16x16x4_f32(false, a, false, b, (short)0,
                                                    acc, false, false);
      }
      eq_epilogue(acc, e0, nt, 1 + m, g_eq_in, active, cn, out_e, hi, ln);
    } else {
      const int nt = wave - 6;
      v8f acc;
      const float bias = beq0[nt * 16 + ln];
#pragma unroll
      for (int i = 0; i < 8; i++) acc[i] = bias;
      for (int kk = 0; kk < 24; kk++) {
        const int k0 = kk * 4 + 2 * hi;
        float2 af = *(const float2*)&tps[ln * TPS_STRIDE + k0];
        float2 bf = *(const float2*)&Weq0[(size_t)(nt * 16 + ln) * 96 + k0];
        v2f a; a.x = af.x; a.y = af.y;
        v2f b; b.x = bf.x; b.y = bf.y;
        acc = __builtin_amdgcn_wmma_f32_16x16x4_f32(false, a, false, b, (short)0,
                                                    acc, false, false);
      }
      eq_epilogue(acc, e0, nt, 0, g_eq_in, active, cn, out_e, hi, ln);
    }
  }
  __syncthreads();

  // ============ phase B: overwrite buffer with l=2 data, mix (10 jobs) ======
#pragma unroll
  for (int p = 0; p < 6; p++)
#pragma unroll
    for (int k = 0; k < 5; k++)
      buf[k * TL_PLANE + wave * TL_STRIDE + p * 32 + lane] = l2v[p][k];
  __syncthreads();

  if (wave < 10) {
    const int kc = wave >> 1, nt = wave & 1;
    const float* Ab = buf + kc * TL_PLANE;
    v8f acc;
#pragma unroll
    for (int i = 0; i < 8; i++) acc[i] = 0.f;
    for (int kk = 0; kk < 48; kk++) {
      const int k0 = kk * 4 + 2 * hi;
      float2 af = *(const float2*)&Ab[ln * TL_STRIDE + k0];
      float2 bf = *(const float2*)&Weq2[(size_t)(nt * 16 + ln) * 192 + k0];
      v2f a; a.x = af.x; a.y = af.y;
      v2f b; b.x = bf.x; b.y = bf.y;
      acc = __builtin_amdgcn_wmma_f32_16x16x4_f32(false, a, false, b, (short)0,
                                                  acc, false, false);
    }
    eq_epilogue(acc, e0, nt, 4 + kc, g_eq_in, active, cn, out_e, hi, ln);
  }
}

// ---- kernel 3b: scalar MLP via V_WMMA_F32_16X16X4_F32 (fp32-exact) ----
#define XS_STRIDE 132  // 128 + 4 pad: conflict-free A-frag reads
__global__ __launch_bounds__(256) void k_scalar_mlp(
    const float* __restrict__ Xin, const float* __restrict__ W1,
    const float* __restrict__ b1, const float* __restrict__ W2,
    const float* __restrict__ b2, const float* __restrict__ uc,
    const int* __restrict__ active, float* __restrict__ out_s) {
  __shared__ float Xs[32 * XS_STRIDE];
  __shared__ float Hs[32 * XS_STRIDE];
  const int tid = threadIdx.x;
  const int e0 = blockIdx.x * 32;
  const float4* src = (const float4*)(Xin + (size_t)e0 * LAT);
  for (int i = tid; i < 32 * LAT / 4; i += 256) {
    float4 v = src[i];
    const int row = i >> 5, c4 = (i & 31) * 4;
    float* d = &Xs[row * XS_STRIDE + c4];
    d[0] = v.x; d[1] = v.y; d[2] = v.z; d[3] = v.w;
  }
  if (tid < 128) {
    __builtin_prefetch(&W1[tid * LAT], 0, 0);
    __builtin_prefetch(&W2[tid * LAT], 0, 0);
  }
  __syncthreads();

  const int wave = tid >> 5, lane = tid & 31;
  const int hi = lane >> 4, ln = lane & 15;
  const float c = uc[0];
  const float cn = c * rsqrtf(c * c + 1.f);

  // layer 1: H = silu(X @ W1^T + b1)
#pragma unroll
  for (int tt = 0; tt < 2; tt++) {
    const int t = wave + tt * 8;
    const int m0 = (t >> 3) * 16, n0 = (t & 7) * 16;
    v8f acc;
    const float bias = b1[n0 + ln];
#pragma unroll
    for (int i = 0; i < 8; i++) acc[i] = bias;
    for (int kk = 0; kk < 32; kk++) {
      const int k0 = kk * 4 + 2 * hi;
      float2 af = *(const float2*)&Xs[(m0 + ln) * XS_STRIDE + k0];
      float2 bf = *(const float2*)&W1[(size_t)(n0 + ln) * LAT + k0];
      v2f a; a.x = af.x; a.y = af.y;
      v2f b; b.x = bf.x; b.y = bf.y;
      acc = __builtin_amdgcn_wmma_f32_16x16x4_f32(false, a, false, b, (short)0,
                                                  acc, false, false);
    }
#pragma unroll
    for (int i = 0; i < 8; i++) {
      float x = acc[i];
      Hs[(m0 + i + 8 * hi) * XS_STRIDE + n0 + ln] = x / (1.f + expf(-x));  // silu
    }
  }
  __syncthreads();

  // layer 2: Y = H @ W2^T + b2; residual add into out_s[active]
#pragma unroll
  for (int tt = 0; tt < 2; tt++) {
    const int t = wave + tt * 8;
    const int m0 = (t >> 3) * 16, n0 = (t & 7) * 16;
    v8f acc;
    const float bias = b2[n0 + ln];
#pragma unroll
    for (int i = 0; i < 8; i++) acc[i] = bias;
    for (int kk = 0; kk < 32; kk++) {
      const int k0 = kk * 4 + 2 * hi;
      float2 af = *(const float2*)&Hs[(m0 + ln) * XS_STRIDE + k0];
      float2 bf = *(const float2*)&W2[(size_t)(n0 + ln) * LAT + k0];
      v2f a; a.x = af.x; a.y = af.y;
      v2f b; b.x = bf.x; b.y = bf.y;
      acc = __builtin_amdgcn_wmma_f32_16x16x4_f32(false, a, false, b, (short)0,
                                                  acc, false, false);
    }
#pragma unroll
    for (int i = 0; i < 8; i++) {
      const int r = m0 + i + 8 * hi;
      const int ae = active[e0 + r];
      float* dp = out_s + (size_t)ae * LAT + n0 + ln;
      *dp += cn * acc[i];
    }
  }
}

extern "C" void kernel_launch(void* const* d_in, const int* in_sizes, int n_in,
                              void* d_out, int out_size, void* d_ws, size_t ws_size,
                              hipStream_t stream) {
  (void)in_sizes; (void)n_in; (void)out_size; (void)ws_size;
  const float* scal   = (const float*)d_in[0];
  const float* eq     = (const float*)d_in[1];
  const float* cond   = (const float*)d_in[2];
  const float* uc     = (const float*)d_in[3];
  const float* W_env  = (const float*)d_in[4];
  const float* b_env0 = (const float*)d_in[5];
  const float* gamma_env = (const float*)d_in[6];
  const float* Wg_env = (const float*)d_in[7];
  const float* bg_env = (const float*)d_in[8];
  const float* gamma_tp = (const float*)d_in[9];
  const float* W1_s   = (const float*)d_in[10];
  const float* b1_s   = (const float*)d_in[11];
  const float* W2_s   = (const float*)d_in[12];
  const float* b2_s   = (const float*)d_in[13];
  const float* Weq0   = (const float*)d_in[14];
  const float* beq0   = (const float*)d_in[15];
  const float* Weq1   = (const float*)d_in[16];
  const float* Weq2   = (const float*)d_in[17];
  const float* Wg_eq  = (const float*)d_in[18];
  const float* bg_eq  = (const float*)d_in[19];
  const int* edge_index = (const int*)d_in[20];
  const int* active   = (const int*)d_in[21];

  float* out = (float*)d_out;
  float* out_s = out;
  float* out_e = out + (size_t)E_EDGES * LAT;

  float* env_nodes = (float*)d_ws;                         // N*288 f32
  float* g_eq = env_nodes + (size_t)N_NODES * MUL * 9;     // E f32
  float* sin_ws = g_eq + E_EDGES;                          // E*128 f32

  const int n_env = N_NODES * MUL * 9;
  k_zero<<<(n_env + 255) / 256, 256, 0, stream>>>(env_nodes, n_env);

  const int n_res = E_EDGES * LAT + E_EDGES * MUL * 9;
  k_resbase<<<(n_res + 255) / 256, 256, 0, stream>>>(scal, eq, uc, out);

  k_env_scatter<<<E_EDGES / 8, 256, 0, stream>>>(scal, eq, cond, W_env, b_env0,
                                                 Wg_env, bg_env, Wg_eq, bg_eq,
                                                 edge_index, env_nodes, g_eq);

  k_node_ln<<<N_NODES / 8, 256, 0, stream>>>(env_nodes, gamma_env);

  k_tp_mix<<<E_EDGES / 16, 512, 0, stream>>>(eq, env_nodes, cond, edge_index,
                                             gamma_tp, Weq0, beq0, Weq1, Weq2,
                                             g_eq, uc, active, sin_ws, out_e);

  k_scalar_mlp<<<E_EDGES / 32, 256, 0, stream>>>(sin_ws, W1_s, b1_s, W2_s, b2_s,
                                                 uc, active, out_s);
}